// SpectralConv2d_69896297775483
// MI455X (gfx1250) — compile-verified
//
#include <hip/hip_runtime.h>
#include <hip/hip_bf16.h>

typedef float v2f __attribute__((ext_vector_type(2)));
typedef float v8f __attribute__((ext_vector_type(8)));

static __device__ __forceinline__ v8f wmma4(v2f a, v2f b, v8f c) {
  // D = A(16x4 f32) * B(4x16 f32) + C(16x16 f32)
  return __builtin_amdgcn_wmma_f32_16x16x4_f32(false, a, false, b, (short)0, c,
                                               false, false);
}

// ---------------------------------------------------------------------------
// Workspace layout (floats):
//   A1  [64][256]   forward row-mode matrix        off 0       (16384)
//   A2t [256][32]   forward col-mode matrix (w,c)  off 16384   (8192)
//   B1  [256][64]   inverse row matrix (h,r)       off 24576   (16384)
//   B2  [32][256]   inverse col matrix (c,w)       off 40960   (8192)
//   Y2  [B*Cin][64][32]                            off 49152   (1048576)
//   FT  [B][Cout][64][32]                          off 1097728 (1048576)
// ---------------------------------------------------------------------------
#define OFF_A1  0
#define OFF_A2T 16384
#define OFF_B1  24576
#define OFF_B2  40960
#define OFF_Y2  49152
#define OFF_FT  1097728

__global__ __launch_bounds__(256) void init_tables(float* __restrict__ ws) {
  const double PI = 3.14159265358979323846;
  int idx = blockIdx.x * 256 + threadIdx.x;
  if (idx < 16384) {                       // A1[r][h]
    int r = idx >> 8, h = idx & 255;
    int rho = (r < 32) ? r : r + 192;
    float wt = (h == 0 || h == 255) ? 1.0f : 2.0f;
    ws[OFF_A1 + idx] = wt * (float)cos(PI * (double)(h * rho) / 255.0);
  } else if (idx < 24576) {                // A2t[w][c]
    int t = idx - 16384;
    int w = t >> 5, c = t & 31;
    float wt = (w == 0 || w == 255) ? 1.0f : 2.0f;
    ws[idx] = wt * (float)cos(PI * (double)(w * c) / 255.0);
  } else if (idx < 40960) {                // B1[h][r]
    int t = idx - 24576;
    int h = t >> 6, r = t & 63;
    int rho = (r < 32) ? r : r + 192;
    float wt = (r == 0 || r == 63) ? 1.0f : 2.0f;  // rho==0 or rho==255
    ws[idx] = wt * (float)cos(PI * (double)(h * rho) / 255.0);
  } else if (idx < 49152) {                // B2[c][w]
    int t = idx - 40960;
    int c = t >> 8, w = t & 255;
    float wt = (c == 0) ? 1.0f : 2.0f;
    ws[idx] = wt * (float)cos(PI * (double)(w * c) / 255.0);
  }
}

// ---------------------------------------------------------------------------
// Stage A: per input image (b,i): Y2 = A1 (64x256) @ X (256x256) @ A2^T (256x32)
// One block per image, 8 waves. GEMM1 in two 128-column chunks via LDS.
// ---------------------------------------------------------------------------
__global__ __launch_bounds__(256) void spectral_fwd(
    const float* __restrict__ x, const float* __restrict__ A1,
    const float* __restrict__ A2t, float* __restrict__ y2out) {
  __shared__ float y1[64 * 132];           // 33 KiB, stride 132 (bank-safe)
  const int img  = blockIdx.x;             // b*64 + i
  const float* X = x + (size_t)img * 65536;
  const int wave = threadIdx.x >> 5;
  const int lane = threadIdx.x & 31;
  const int lo = lane & 15, hi = lane >> 4;
  const int mt = wave & 3;                 // M-tile (both GEMMs)
  const int ng = wave >> 2;                // N group (GEMM1) / N-tile (GEMM2)

  v8f accY = {};                           // GEMM2 accumulator: y2 tile (mt,ng)
  for (int chunk = 0; chunk < 2; ++chunk) {
    // ---- GEMM1: y1[64][128] = A1 @ X[:, chunk*128 : chunk*128+128] ----
    v8f acc[4];
#pragma unroll
    for (int t = 0; t < 4; ++t) acc[t] = (v8f){};
    const int wbase = chunk * 128 + ng * 64;
    for (int k0 = 0; k0 < 256; k0 += 4) {
      const int kr = k0 + 2 * hi;
      v2f a;
      a.x = A1[(mt * 16 + lo) * 256 + kr];
      a.y = A1[(mt * 16 + lo) * 256 + kr + 1];
#pragma unroll
      for (int t = 0; t < 4; ++t) {
        const int wc = wbase + t * 16 + lo;
        v2f bf;
        bf.x = X[kr * 256 + wc];
        bf.y = X[(kr + 1) * 256 + wc];
        acc[t] = wmma4(a, bf, acc[t]);
      }
    }
    __syncthreads();                       // previous GEMM2 reads are done
#pragma unroll
    for (int t = 0; t < 4; ++t) {
      const int col = ng * 64 + t * 16 + lo;
#pragma unroll
      for (int v = 0; v < 8; ++v)
        y1[(mt * 16 + v + 8 * hi) * 132 + col] = acc[t][v];
    }
    __syncthreads();
    // ---- GEMM2 partial: y2 += y1(64x128) @ A2t-chunk ----
    for (int k0 = 0; k0 < 128; k0 += 4) {
      const int kk = k0 + 2 * hi;
      v2f a, bf;
      a.x = y1[(mt * 16 + lo) * 132 + kk];
      a.y = y1[(mt * 16 + lo) * 132 + kk + 1];
      const int c = ng * 16 + lo;
      bf.x = A2t[(chunk * 128 + kk) * 32 + c];
      bf.y = A2t[(chunk * 128 + kk + 1) * 32 + c];
      accY = wmma4(a, bf, accY);
    }
  }
  float* Y = y2out + (size_t)img * 2048;   // [64][32]
#pragma unroll
  for (int v = 0; v < 8; ++v)
    Y[(mt * 16 + v + 8 * hi) * 32 + ng * 16 + lo] = accY[v];
}

// ---------------------------------------------------------------------------
// Stage B: per mode (m1,m2): FT[b][o] = sum_i Y2[b][i] * W[i][o][m1e][m2]
// 8 waves / block, one mode per wave; 8x64x64 GEMM on a 16x16 WMMA (rows 8..15
// padded with zeros and discarded).
// ---------------------------------------------------------------------------
__global__ __launch_bounds__(256) void spectral_mix(
    const float* __restrict__ y2, const float* __restrict__ w1,
    const float* __restrict__ w2, float* __restrict__ ft) {
  const int wave = threadIdx.x >> 5;
  const int lane = threadIdx.x & 31;
  const int lo = lane & 15, hi = lane >> 4;
  const int mode = blockIdx.x * 8 + wave;
  const int m1 = mode >> 5, m2 = mode & 31;
  const float* wsel = (m1 < 32) ? w1 : w2;
  const int woff = ((m1 < 32) ? m1 : (m1 - 32)) * 32 + m2;

  v8f acc[4];
#pragma unroll
  for (int t = 0; t < 4; ++t) acc[t] = (v8f){};
  for (int k0 = 0; k0 < 64; k0 += 4) {
    const int kk = k0 + 2 * hi;
    v2f a = {0.0f, 0.0f};
    if (lo < 8) {                          // batch rows 0..7 live, rest zero
      a.x = y2[(size_t)(lo * 64 + kk) * 2048 + m1 * 32 + m2];
      a.y = y2[(size_t)(lo * 64 + kk + 1) * 2048 + m1 * 32 + m2];
    }
#pragma unroll
    for (int t = 0; t < 4; ++t) {
      const int o = t * 16 + lo;
      v2f bf;
      bf.x = wsel[(size_t)(kk * 64 + o) * 1024 + woff];
      bf.y = wsel[(size_t)((kk + 1) * 64 + o) * 1024 + woff];
      acc[t] = wmma4(a, bf, acc[t]);
    }
  }
  if (hi == 0) {                           // rows b = v (0..7)
#pragma unroll
    for (int t = 0; t < 4; ++t)
#pragma unroll
      for (int v = 0; v < 8; ++v) {
        const int o = t * 16 + lo;
        ft[(size_t)(v * 64 + o) * 2048 + m1 * 32 + m2] = acc[t][v];
      }
  }
}

// ---------------------------------------------------------------------------
// Stage C: per output image (b,o): out = B1(256x64) @ FT(64x32) @ B2(32x256)
// One block per image, 8 waves; FT and T staged in LDS.
// ---------------------------------------------------------------------------
__global__ __launch_bounds__(256) void spectral_inv(
    const float* __restrict__ ft, const float* __restrict__ B1,
    const float* __restrict__ B2, float* __restrict__ out) {
  __shared__ float ftl[64 * 33];           // 8.25 KiB
  __shared__ float Tl[256 * 36];           // 36 KiB
  const int img = blockIdx.x;              // b*64 + o
  const int tid = threadIdx.x;
  const int wave = tid >> 5;
  const int lane = tid & 31;
  const int lo = lane & 15, hi = lane >> 4;

  const float* F = ft + (size_t)img * 2048;
  for (int idx = tid; idx < 2048; idx += 256)
    ftl[(idx >> 5) * 33 + (idx & 31)] = F[idx];
  __syncthreads();

  // ---- C1: T[256][32] = B1 @ ftl ----
#pragma unroll
  for (int t = 0; t < 4; ++t) {
    const int tile = wave * 4 + t;
    const int mt = tile >> 1, nt = tile & 1;
    v8f acc = {};
    const int m = mt * 16 + lo;
    for (int k0 = 0; k0 < 64; k0 += 4) {
      const int kk = k0 + 2 * hi;
      v2f a, bf;
      a.x = B1[m * 64 + kk];
      a.y = B1[m * 64 + kk + 1];
      bf.x = ftl[kk * 33 + nt * 16 + lo];
      bf.y = ftl[(kk + 1) * 33 + nt * 16 + lo];
      acc = wmma4(a, bf, acc);
    }
#pragma unroll
    for (int v = 0; v < 8; ++v)
      Tl[(mt * 16 + v + 8 * hi) * 36 + nt * 16 + lo] = acc[v];
  }
  __syncthreads();

  // ---- C2: out = T @ B2 ----
  float* O = out + (size_t)img * 65536;
#pragma unroll
  for (int mtl = 0; mtl < 2; ++mtl) {
    const int mt = wave * 2 + mtl;
    v2f af[8];
#pragma unroll
    for (int ks = 0; ks < 8; ++ks) {
      const int kk = ks * 4 + 2 * hi;
      af[ks].x = Tl[(mt * 16 + lo) * 36 + kk];
      af[ks].y = Tl[(mt * 16 + lo) * 36 + kk + 1];
    }
    for (int nt = 0; nt < 16; ++nt) {
      v8f acc = {};
#pragma unroll
      for (int ks = 0; ks < 8; ++ks) {
        const int kk = ks * 4 + 2 * hi;
        v2f bf;
        bf.x = B2[kk * 256 + nt * 16 + lo];
        bf.y = B2[(kk + 1) * 256 + nt * 16 + lo];
        acc = wmma4(af[ks], bf, acc);
      }
#pragma unroll
      for (int v = 0; v < 8; ++v)
        O[(size_t)(mt * 16 + v + 8 * hi) * 256 + nt * 16 + lo] = acc[v];
    }
  }
}

extern "C" void kernel_launch(void* const* d_in, const int* in_sizes, int n_in,
                              void* d_out, int out_size, void* d_ws,
                              size_t ws_size, hipStream_t stream) {
  (void)in_sizes; (void)n_in; (void)out_size; (void)ws_size;
  const float* x  = (const float*)d_in[0];   // [8][64][256][256]
  const float* w1 = (const float*)d_in[1];   // [64][64][32][32]
  const float* w2 = (const float*)d_in[2];   // [64][64][32][32]
  float* out = (float*)d_out;                // [8][64][256][256]
  float* ws  = (float*)d_ws;                 // needs ~8.2 MiB

  float* A1  = ws + OFF_A1;
  float* A2t = ws + OFF_A2T;
  float* B1  = ws + OFF_B1;
  float* B2  = ws + OFF_B2;
  float* Y2  = ws + OFF_Y2;
  float* FT  = ws + OFF_FT;

  init_tables<<<192, 256, 0, stream>>>(ws);
  spectral_fwd<<<512, 256, 0, stream>>>(x, A1, A2t, Y2);
  spectral_mix<<<256, 256, 0, stream>>>(Y2, w1, w2, FT);
  spectral_inv<<<512, 256, 0, stream>>>(FT, B1, B2, out);
}